// CustomMaskRCNN_12678743457975
// MI455X (gfx1250) — compile-verified
//
#include <hip/hip_runtime.h>

// Segment-max over detection masks:
//   out[i, c, p] = max(0, max_{j : labels[i,j]==c} masks[i, j, p])
//
// Bandwidth-bound: ~613 MB streamed once => ~26 us floor at 23.3 TB/s.
// Max-semiring reduction has no WMMA formulation; the gfx1250-specific wins
// are wave32 block sizing, B128 non-temporal loads/stores (working set
// 613 MB >> 192 MB L2), global_prefetch_b8 for the 196 KB inter-detection
// strides, and GLOBAL_LOAD_ASYNC_TO_LDS_B128 (+ s_wait_asynccnt) to stage
// each block's detection list into LDS once instead of re-reading it from
// global per iteration per wave.

typedef float v4f __attribute__((ext_vector_type(4)));

#define NIMG 16
#define NDET 100
#define NDET_PAD 128        // list stride: 128 u16 = 256 B = 16 lanes x B128
#define NCLS 91
#define HW   (224 * 224)    // 50176 pixels per plane
#define HW4  (HW / 4)       // 12544 float4s per plane (= 49 * 256 exactly)

// ---------------------------------------------------------------------------
// Kernel 1: build per-(image, class) detection index lists.
// One thread per (image, class); scans the 100 labels of its image.
// ---------------------------------------------------------------------------
__global__ void build_lists_kernel(const int* __restrict__ labels,
                                   int* __restrict__ counts,
                                   unsigned short* __restrict__ lists) {
    const int idx = blockIdx.x * blockDim.x + threadIdx.x;
    if (idx >= NIMG * NCLS) return;
    const int img = idx / NCLS;
    const int cls = idx - img * NCLS;
    const int* lab = labels + img * NDET;
    unsigned short* lst = lists + idx * NDET_PAD;
    int cnt = 0;
    #pragma unroll 4
    for (int j = 0; j < NDET; ++j) {
        if (lab[j] == cls) {
            lst[cnt] = (unsigned short)j;
            ++cnt;
        }
    }
    counts[idx] = cnt;
}

// ---------------------------------------------------------------------------
// Kernel 2: gather-max. Grid (49 tiles, 91 classes, 16 images), block 256
// (= 8 wave32s). The block's detection list (256 B) is staged into LDS with
// one async B128 burst; each thread then owns one float4 of the output plane
// and max-reduces over that class's detection planes with a 1-deep software
// pipeline + prefetch of the following detection's cachelines.
// ---------------------------------------------------------------------------
__global__ void __launch_bounds__(256)
segmax_kernel(const float* __restrict__ masks,
              const int* __restrict__ counts,
              const unsigned short* __restrict__ lists,
              float* __restrict__ out) {
    const int tile = blockIdx.x;                       // 0..48
    const int cls  = blockIdx.y;                       // 0..90
    const int img  = blockIdx.z;                       // 0..15
    const int seg  = img * NCLS + cls;                 // 0..1455
    const int p4   = tile * 256 + threadIdx.x;         // float4 index in plane

    __shared__ unsigned short s_lst[NDET_PAD];         // 256 B staged list

    // Stage the whole padded list with 16 lanes x GLOBAL_LOAD_ASYNC_TO_LDS_B128.
    // LDS byte address = low 32 bits of the generic pointer (LDS aperture).
    if (threadIdx.x < 16) {
        unsigned lds_addr = (unsigned)(uintptr_t)(&s_lst[0]) + threadIdx.x * 16u;
        unsigned long long gaddr =
            (unsigned long long)(uintptr_t)(lists + (size_t)seg * NDET_PAD) +
            threadIdx.x * 16ull;
        asm volatile("global_load_async_to_lds_b128 %0, %1, off"
                     :
                     : "v"(lds_addr), "v"(gaddr)
                     : "memory");
        asm volatile("s_wait_asynccnt 0x0" ::: "memory");
    }
    __syncthreads();

    const int cnt = counts[seg];

    // Base of this image's mask planes, offset to this thread's float4 slot.
    const v4f* __restrict__ mbase =
        (const v4f*)masks + (size_t)img * NDET * HW4 + p4;

    v4f acc = (v4f){0.f, 0.f, 0.f, 0.f};   // zero init == clamp max(out, 0)

    if (cnt > 0) {
        // Software pipeline: keep one non-temporal B128 load in flight and
        // prefetch the detection after that (planes are 196 KB apart, so
        // each iteration jumps to cold cachelines -> global_prefetch_b8).
        int j0 = s_lst[0];
        v4f v = __builtin_nontemporal_load(mbase + (size_t)j0 * HW4);
        for (int k = 1; k < cnt; ++k) {
            const v4f* srcn = mbase + (size_t)s_lst[k] * HW4;
            if (k + 1 < cnt) {
                __builtin_prefetch(mbase + (size_t)s_lst[k + 1] * HW4, 0, 0);
            }
            v4f vn = __builtin_nontemporal_load(srcn);
            acc.x = fmaxf(acc.x, v.x);
            acc.y = fmaxf(acc.y, v.y);
            acc.z = fmaxf(acc.z, v.z);
            acc.w = fmaxf(acc.w, v.w);
            v = vn;
        }
        acc.x = fmaxf(acc.x, v.x);
        acc.y = fmaxf(acc.y, v.y);
        acc.z = fmaxf(acc.z, v.z);
        acc.w = fmaxf(acc.w, v.w);
    }

    // Write-once stream: non-temporal B128 store.
    v4f* __restrict__ dst = (v4f*)out + (size_t)seg * HW4 + p4;
    __builtin_nontemporal_store(acc, dst);
}

// ---------------------------------------------------------------------------
// Launch wrapper
//   d_in[0]: masks  float32 [16, 100, 224, 224]
//   d_in[1]: labels int     [16, 100]
//   d_out  : float32 [16, 91, 224, 224]
//   d_ws   : counts (int[1456]) + lists (u16[1456][128])  (~380 KB)
// ---------------------------------------------------------------------------
extern "C" void kernel_launch(void* const* d_in, const int* in_sizes, int n_in,
                              void* d_out, int out_size, void* d_ws, size_t ws_size,
                              hipStream_t stream) {
    const float* masks  = (const float*)d_in[0];
    const int*   labels = (const int*)d_in[1];
    float* out = (float*)d_out;

    char* ws = (char*)d_ws;
    int* counts = (int*)ws;
    size_t lists_off = ((size_t)(NIMG * NCLS) * sizeof(int) + 255) & ~(size_t)255;
    unsigned short* lists = (unsigned short*)(ws + lists_off);

    {
        const int total = NIMG * NCLS;            // 1456
        dim3 grid((total + 127) / 128), block(128);
        build_lists_kernel<<<grid, block, 0, stream>>>(labels, counts, lists);
    }
    {
        dim3 grid(HW4 / 256, NCLS, NIMG);         // (49, 91, 16)
        dim3 block(256);                          // 8 wave32s
        segmax_kernel<<<grid, block, 0, stream>>>(masks, counts, lists, out);
    }
}